// MeshEncoder_13649406066714
// MI455X (gfx1250) — compile-verified
//
#include <hip/hip_runtime.h>
#include <hip/hip_bf16.h>

typedef float v2f __attribute__((ext_vector_type(2)));
typedef float v8f __attribute__((ext_vector_type(8)));

#define HID 64

// ---------- degree / normalization ----------
__global__ void k_deg_init(float* __restrict__ deg, int n) {
    int i = blockIdx.x * blockDim.x + threadIdx.x;
    if (i < n) deg[i] = 1.0f;               // self-loop contribution
}

__global__ void k_deg_accum(const long long* __restrict__ dst,
                            float* __restrict__ deg, int nE) {
    int e = blockIdx.x * blockDim.x + threadIdx.x;
    if (e < nE) atomicAdd(&deg[(int)dst[e]], 1.0f);
}

__global__ void k_deg_finalize(float* __restrict__ dis, int n) {
    int i = blockIdx.x * blockDim.x + threadIdx.x;
    if (i < n) dis[i] = rsqrtf(dis[i]);     // deg >= 1 always
}

// ---------- dense GEMM (WMMA f32 16x16x4), epilogue scales row by dis ----------
// G[row, 0:64] = dis[row] * (X[row, 0:K] @ W[K, 0:64])
template <int KDIM>
__global__ void __launch_bounds__(256)
k_gemm_scale(const float* __restrict__ Xin,
             const float* __restrict__ W,      // KDIM x 64 row-major
             const float* __restrict__ dis,
             float* __restrict__ G,
             int n_nodes) {
    __shared__ float sW[KDIM * HID];
    for (int i = threadIdx.x; i < KDIM * HID; i += 256) sW[i] = W[i];
    __syncthreads();

    const int lane = threadIdx.x & 31;
    const int wave = threadIdx.x >> 5;               // 8 waves / block
    const long row_base = (long)blockIdx.x * 128 + (long)wave * 16;
    if (row_base >= n_nodes) return;                 // n_nodes % 16 == 0 -> whole-wave tiles

    const int l15  = lane & 15;                      // M (for A) / N (for B,C,D)
    const int koff = (lane >> 4) * 2;                // lanes 16-31 hold K+2,K+3

    v8f acc0 = {}, acc1 = {}, acc2 = {}, acc3 = {};
    const float* Xrow = Xin + (row_base + l15) * KDIM;

    for (int k0 = 0; k0 < KDIM; k0 += 4) {
        v2f a;
        a.x = Xrow[k0 + koff];
        a.y = Xrow[k0 + koff + 1];
        const float* wp0 = &sW[(k0 + koff) * HID];
        const float* wp1 = &sW[(k0 + koff + 1) * HID];
        v2f b0, b1, b2, b3;
        b0.x = wp0[l15];       b0.y = wp1[l15];
        b1.x = wp0[16 + l15];  b1.y = wp1[16 + l15];
        b2.x = wp0[32 + l15];  b2.y = wp1[32 + l15];
        b3.x = wp0[48 + l15];  b3.y = wp1[48 + l15];
        acc0 = __builtin_amdgcn_wmma_f32_16x16x4_f32(false, a, false, b0, (short)0, acc0, false, false);
        acc1 = __builtin_amdgcn_wmma_f32_16x16x4_f32(false, a, false, b1, (short)0, acc1, false, false);
        acc2 = __builtin_amdgcn_wmma_f32_16x16x4_f32(false, a, false, b2, (short)0, acc2, false, false);
        acc3 = __builtin_amdgcn_wmma_f32_16x16x4_f32(false, a, false, b3, (short)0, acc3, false, false);
    }

    // C/D layout: VGPR r -> row row_base+r (lanes 0-15) / row_base+r+8 (lanes 16-31), col = l15 + 16*ntile
    const int rofs = (lane >> 4) * 8;
#pragma unroll
    for (int r = 0; r < 8; ++r) {
        const long row = row_base + r + rofs;
        const float s = dis[row];
        float* o = G + row * HID + l15;
        o[0]  = s * acc0[r];
        o[16] = s * acc1[r];
        o[32] = s * acc2[r];
        o[48] = s * acc3[r];
    }
}

// ---------- zero the scatter accumulator ----------
__global__ void k_zero4(float4* __restrict__ p, long n4) {
    long i = (long)blockIdx.x * blockDim.x + threadIdx.x;
    if (i < n4) p[i] = make_float4(0.f, 0.f, 0.f, 0.f);
}

// ---------- edge scatter: S[dst] += G[src] ----------
__global__ void k_scatter(const long long* __restrict__ src,
                          const long long* __restrict__ dst,
                          const float* __restrict__ G,
                          float* __restrict__ S,
                          long nE) {
    long tid = (long)blockIdx.x * blockDim.x + threadIdx.x;
    const long total = nE * HID;
    if (tid >= total) return;
    const long e = tid >> 6;                 // HID == 64
    const int  c = (int)(tid & 63);
    const long s = (long)src[e];
    const long d = (long)dst[e];
    atomicAdd(&S[d * HID + c], G[s * HID + c]);
}

// ---------- finalize: out = [relu](dis[i]*(G[i]+S[i]) + b) ----------
__global__ void k_finalize(const float* __restrict__ G,
                           const float* __restrict__ S,
                           const float* __restrict__ dis,
                           const float* __restrict__ bias,
                           float* __restrict__ out,
                           long n, int do_relu) {
    long tid = (long)blockIdx.x * blockDim.x + threadIdx.x;
    const long total = n * HID;
    if (tid >= total) return;
    const long i = tid >> 6;
    const int  c = (int)(tid & 63);
    float v = dis[i] * (G[tid] + S[tid]) + bias[c];
    if (do_relu) v = fmaxf(v, 0.0f);
    out[tid] = v;
}

extern "C" void kernel_launch(void* const* d_in, const int* in_sizes, int n_in,
                              void* d_out, int out_size, void* d_ws, size_t ws_size,
                              hipStream_t stream) {
    (void)n_in; (void)out_size; (void)ws_size;
    const float*      x   = (const float*)d_in[0];      // [N,32]
    const long long*  ei  = (const long long*)d_in[1];  // [2,E] int64
    const float*      W1  = (const float*)d_in[2];
    const float*      b1  = (const float*)d_in[3];
    const float*      W2  = (const float*)d_in[4];
    const float*      b2  = (const float*)d_in[5];
    const float*      W3  = (const float*)d_in[6];
    const float*      b3  = (const float*)d_in[7];
    float*            out = (float*)d_out;

    const int  N  = in_sizes[0] / 32;       // 100000
    const long NE = (long)(in_sizes[1] / 2);// 1600000
    const long NF = (long)N * HID;          // node-feature floats

    const long long* e_src = ei;
    const long long* e_dst = ei + NE;

    float* ws  = (float*)d_ws;
    float* dis = ws;                        // N floats (padded region 102400)
    float* G   = ws + 102400;               // NF floats
    float* S   = G + NF;                    // NF floats
    float* A   = S + NF;                    // NF floats

    const int T = 256;
    const int gN   = (N + T - 1) / T;
    const int gE   = (int)((NE + T - 1) / T);
    const int gEF  = (int)((NE * HID + T - 1) / T);
    const int gNF  = (int)((NF + T - 1) / T);
    const int gNF4 = (int)((NF / 4 + T - 1) / T);
    const int gGEMM = (N + 127) / 128;      // 128 rows per block (8 waves x 16)

    // ---- normalization ----
    k_deg_init<<<gN, T, 0, stream>>>(dis, N);
    k_deg_accum<<<gE, T, 0, stream>>>(e_dst, dis, (int)NE);
    k_deg_finalize<<<gN, T, 0, stream>>>(dis, N);

    // ---- layer 1: X(32) -> A ----
    k_gemm_scale<32><<<gGEMM, T, 0, stream>>>(x, W1, dis, G, N);
    k_zero4<<<gNF4, T, 0, stream>>>((float4*)S, NF / 4);
    k_scatter<<<gEF, T, 0, stream>>>(e_src, e_dst, G, S, NE);
    k_finalize<<<gNF, T, 0, stream>>>(G, S, dis, b1, A, N, 1);

    // ---- layer 2: A(64) -> A ----
    k_gemm_scale<64><<<gGEMM, T, 0, stream>>>(A, W2, dis, G, N);
    k_zero4<<<gNF4, T, 0, stream>>>((float4*)S, NF / 4);
    k_scatter<<<gEF, T, 0, stream>>>(e_src, e_dst, G, S, NE);
    k_finalize<<<gNF, T, 0, stream>>>(G, S, dis, b2, A, N, 1);

    // ---- layer 3: A(64) -> out (no relu) ----
    k_gemm_scale<64><<<gGEMM, T, 0, stream>>>(A, W3, dis, G, N);
    k_zero4<<<gNF4, T, 0, stream>>>((float4*)S, NF / 4);
    k_scatter<<<gEF, T, 0, stream>>>(e_src, e_dst, G, S, NE);
    k_finalize<<<gNF, T, 0, stream>>>(G, S, dis, b3, out, N, 0);
}